// se3D_14894946582876
// MI455X (gfx1250) — compile-verified
//
#include <hip/hip_runtime.h>
#include <math.h>

typedef __attribute__((ext_vector_type(2))) float v2f;
typedef __attribute__((ext_vector_type(4))) float v4f;
typedef __attribute__((ext_vector_type(8))) float v8f;

#define CC    64
#define MSPAT (31 * 64 * 64)      // 126976 spatial elements per (b,c)
#define NB    8                   // batch
#define NKB   62                  // K-blocks per batch (62 * 32 * 64 = 126976)
#define LDP   65                  // LDS row stride for k_small (ds path)
#define LDG   68                  // LDS row stride for k_gram: 272B rows -> 16B aligned
                                  // for async B128 stores; 68%64=4 -> conflict-free rows

// ---------------------------------------------------------------------------
// Kernel 1: partial Gram (X X^T) + partial row-sums per (batch, k-block).
// 256 threads = 8 waves; 16 output tiles of 16x16 -> 2 tiles per wave.
// Global -> LDS staging uses CDNA5 GLOBAL_LOAD_ASYNC_TO_LDS_B128 with a
// double-buffered LDS chunk: chunk ch+1 streams in (ASYNCcnt) while WMMAs
// consume chunk ch. No VGPR round-trip, no ds_store traffic.
// ---------------------------------------------------------------------------
__global__ __launch_bounds__(256) void k_gram(const float* __restrict__ x,
                                              float* __restrict__ pgram,
                                              float* __restrict__ psum) {
    __shared__ float Xs[2 * 64 * LDG];
    const int tid  = threadIdx.x;
    const int kb   = blockIdx.x;      // 0..61
    const int b    = blockIdx.y;      // 0..7
    const int wave = tid >> 5;
    const int lane = tid & 31;
    const int n    = lane & 15;       // column / row-in-tile
    const int kh   = lane >> 4;       // K-pair half (0:k0,k1  1:k2,k3)

    const float* xb = x + (size_t)b * CC * MSPAT;

    const int t0  = wave * 2, t1 = wave * 2 + 1;
    const int mi0 = t0 >> 2, ni0 = t0 & 3;
    const int mi1 = t1 >> 2, ni1 = t1 & 3;

    v8f acc0 = {};
    v8f acc1 = {};
    float rs = 0.0f;

    const unsigned ldsBase = (unsigned)(size_t)(&Xs[0]);   // low 32b = LDS offset
    const int rowA = tid >> 4;        // this thread's first staged row (0..15, 16 thr/row)
    const int c4A  = tid & 15;        // this thread's float4 column within the row

    // issue one 64x64 chunk (1024 B128 transfers, 4 per thread) into buffer `buf`
    auto issue_chunk = [&](int buf, int k0) {
        #pragma unroll
        for (int i = 0; i < 4; ++i) {
            const int row = rowA + i * 16;                  // 4 rows per thread
            const float* gp = xb + (size_t)row * MSPAT + k0 + c4A * 4;
            const unsigned loff = ldsBase +
                (unsigned)((buf * 64 * LDG + row * LDG + c4A * 4) * 4);
            asm volatile("global_load_async_to_lds_b128 %0, %1, off"
                         :: "v"(loff), "v"(gp) : "memory");
        }
    };

    issue_chunk(0, kb * 2048);                              // prologue: chunk 0

    for (int ch = 0; ch < 32; ++ch) {
        const int cur = ch & 1;
        asm volatile("s_wait_asynccnt 0" ::: "memory");     // my async stores landed
        __syncthreads();                                    // everyone's landed; prev
                                                            // compute on other buf done
        if (ch + 1 < 32) issue_chunk(cur ^ 1, kb * 2048 + (ch + 1) * 64);

        const float* Xc = &Xs[cur * 64 * LDG];

        if (tid < 64) {                                     // per-channel partial sums
            float s = 0.0f;
            #pragma unroll 8
            for (int k = 0; k < 64; ++k) s += Xc[tid * LDG + k];
            rs += s;
        }

        const float* a0r = &Xc[(mi0 * 16 + n) * LDG];
        const float* b0r = &Xc[(ni0 * 16 + n) * LDG];
        const float* a1r = &Xc[(mi1 * 16 + n) * LDG];
        const float* b1r = &Xc[(ni1 * 16 + n) * LDG];
        #pragma unroll
        for (int k = 0; k < 64; k += 4) {
            v2f a, bm;
            a.x  = a0r[k + 2 * kh];  a.y  = a0r[k + 2 * kh + 1];
            bm.x = b0r[k + 2 * kh];  bm.y = b0r[k + 2 * kh + 1];
            acc0 = __builtin_amdgcn_wmma_f32_16x16x4_f32(false, a, false, bm,
                                                         (short)0, acc0, false, false);
            a.x  = a1r[k + 2 * kh];  a.y  = a1r[k + 2 * kh + 1];
            bm.x = b1r[k + 2 * kh];  bm.y = b1r[k + 2 * kh + 1];
            acc1 = __builtin_amdgcn_wmma_f32_16x16x4_f32(false, a, false, bm,
                                                         (short)0, acc1, false, false);
        }
    }

    float* pg = pgram + (size_t)(b * NKB + kb) * 4096;
    #pragma unroll
    for (int r = 0; r < 8; ++r) {
        pg[(mi0 * 16 + r + 8 * kh) * 64 + ni0 * 16 + n] = acc0[r];
        pg[(mi1 * 16 + r + 8 * kh) * 64 + ni1 * 16 + n] = acc1[r];
    }
    if (tid < 64) psum[(size_t)(b * NKB + kb) * 64 + tid] = rs;
}

// ---------------------------------------------------------------------------
// 64x64 LDS matmul: Co = A @ B via v_wmma_f32_16x16x4_f32.
// Caller must have synced so A/B are visible; syncs after writing Co.
// ---------------------------------------------------------------------------
__device__ __forceinline__ void mm64(const float* A, const float* Bm, float* Co, int tid) {
    const int wave = tid >> 5, lane = tid & 31;
    const int n = lane & 15, kh = lane >> 4;
    for (int t = wave; t < 16; t += 8) {
        const int mi = t >> 2, ni = t & 3;
        v8f acc = {};
        const float* arow = A + (mi * 16 + n) * LDP;
        #pragma unroll
        for (int k = 0; k < 64; k += 4) {
            v2f av, bv;
            av.x = arow[k + 2 * kh];
            av.y = arow[k + 2 * kh + 1];
            bv.x = Bm[(k + 2 * kh) * LDP + ni * 16 + n];
            bv.y = Bm[(k + 2 * kh + 1) * LDP + ni * 16 + n];
            acc = __builtin_amdgcn_wmma_f32_16x16x4_f32(false, av, false, bv,
                                                        (short)0, acc, false, false);
        }
        #pragma unroll
        for (int r = 0; r < 8; ++r)
            Co[(mi * 16 + r + 8 * kh) * LDP + ni * 16 + n] = acc[r];
    }
    __syncthreads();
}

// ---------------------------------------------------------------------------
// Kernel 2: per-batch reduce partials -> covariance -> Newton-Schulz sqrtm
// -> branch linears -> sigmoid gate. One WG of 256 threads per batch.
// ---------------------------------------------------------------------------
__global__ __launch_bounds__(256) void k_small(
    const float* __restrict__ pgram, const float* __restrict__ psum,
    const float* __restrict__ w1,  const float* __restrict__ bb1,
    const float* __restrict__ w12, const float* __restrict__ bb12,
    const float* __restrict__ w2,  const float* __restrict__ bb2,
    const float* __restrict__ w22, const float* __restrict__ bb22,
    const float* __restrict__ w3,  const float* __restrict__ bb3,
    float* __restrict__ gout) {
    __shared__ float B0[64 * LDP], B1[64 * LDP], B2[64 * LDP], B3[64 * LDP];
    __shared__ float svec[64], vx1[64], vx2[64], tA[64], tB[64], tC[64], tD[64], sc[1];

    const int tid = threadIdx.x;
    const int b   = blockIdx.x;
    const float invM = 1.0f / (float)MSPAT;

    // 1) reduce the 62 partial Grams / sums
    const float* pg = pgram + (size_t)b * NKB * 4096;
    #pragma unroll
    for (int i = 0; i < 16; ++i) {
        const int idx = tid + i * 256;
        float acc = 0.0f;
        for (int kb = 0; kb < NKB; ++kb) acc += pg[kb * 4096 + idx];
        B0[(idx >> 6) * LDP + (idx & 63)] = acc;
    }
    if (tid < 64) {
        const float* ps = psum + (size_t)b * NKB * 64;
        float acc = 0.0f;
        for (int kb = 0; kb < NKB; ++kb) acc += ps[kb * 64 + tid];
        svec[tid] = acc;
    }
    __syncthreads();

    // 2) covariance: G/M - s s^T / M^2
    #pragma unroll
    for (int i = 0; i < 16; ++i) {
        const int idx = tid + i * 256;
        const int r = idx >> 6, c = idx & 63;
        B0[r * LDP + c] = B0[r * LDP + c] * invM - svec[r] * svec[c] * invM * invM;
    }
    __syncthreads();
    if (tid == 0) {
        float tr = 0.0f;
        for (int i = 0; i < 64; ++i) tr += B0[i * LDP + i];
        sc[0] = tr;
    }
    __syncthreads();
    const float normA = sc[0];
    const float rnorm = 1.0f / normA;

    // 3) An = cov/normA in B0 ; ZY = 0.5*(3I - An) in B1
    #pragma unroll
    for (int i = 0; i < 16; ++i) {
        const int idx = tid + i * 256;
        const int r = idx >> 6, c = idx & 63;
        const float an = B0[r * LDP + c] * rnorm;
        B0[r * LDP + c] = an;
        B1[r * LDP + c] = 0.5f * (((r == c) ? 3.0f : 0.0f) - an);
    }
    __syncthreads();
    mm64(B0, B1, B2, tid);                    // Y = An @ ZY, Z = ZY (B1)

    float *Yp = B2, *Zp = B1, *Tp = B0, *Sp = B3;
    for (int it = 0; it < 3; ++it) {
        mm64(Zp, Yp, Tp, tid);                // T = Z@Y
        #pragma unroll
        for (int i = 0; i < 16; ++i) {        // T = 0.5*(3I - T)
            const int idx = tid + i * 256;
            const int r = idx >> 6, c = idx & 63;
            Tp[r * LDP + c] = 0.5f * (((r == c) ? 3.0f : 0.0f) - Tp[r * LDP + c]);
        }
        __syncthreads();
        mm64(Yp, Tp, Sp, tid);                // Ynew = Y@T   (into spare)
        mm64(Tp, Zp, Yp, tid);                // Znew = T@Z   (into old Y)
        float* oldY = Yp; float* oldZ = Zp;
        Yp = Sp; Zp = oldY; Sp = oldZ;        // rotate buffers
    }

    // final: R = 0.5 * sqrt(normA) * (Y @ (3I - Z@Y))
    mm64(Zp, Yp, Tp, tid);
    #pragma unroll
    for (int i = 0; i < 16; ++i) {
        const int idx = tid + i * 256;
        const int r = idx >> 6, c = idx & 63;
        Tp[r * LDP + c] = ((r == c) ? 3.0f : 0.0f) - Tp[r * LDP + c];
    }
    __syncthreads();
    mm64(Yp, Tp, Sp, tid);

    // 4) x2 = column-mean of R (with deferred 0.5*sqrt scaling) ; x1 = s/M
    if (tid < 64) {
        float acc = 0.0f;
        for (int i = 0; i < 64; ++i) acc += Sp[i * LDP + tid];
        vx2[tid] = acc * (0.5f * sqrtf(normA) * (1.0f / 64.0f));
        vx1[tid] = svec[tid] * invM;
    }
    __syncthreads();

    // 5) linears + gate
    if (tid < 64) {
        float a = bb1[tid], d = bb2[tid];
        for (int c = 0; c < 64; ++c) { a += w1[tid * 64 + c] * vx1[c];
                                       d += w2[tid * 64 + c] * vx2[c]; }
        tA[tid] = a; tB[tid] = d;
    }
    __syncthreads();
    if (tid < 64) {
        float a = bb12[tid], d = bb22[tid];
        for (int c = 0; c < 64; ++c) { a += w12[tid * 64 + c] * tA[c];
                                       d += w22[tid * 64 + c] * tB[c]; }
        tC[tid] = a; tD[tid] = d;
    }
    __syncthreads();
    if (tid < 64) {
        float a = bb3[tid];
        for (int c = 0; c < 64; ++c) a += w3[tid * 128 + c] * tC[c];
        for (int c = 0; c < 64; ++c) a += w3[tid * 128 + 64 + c] * tD[c];
        gout[b * 64 + tid] = 1.0f / (1.0f + expf(-a));
    }
}

// ---------------------------------------------------------------------------
// Kernel 3: out[b,c,:] = g[b,c] * x[b,c,:] (pure bandwidth, clang float4).
// Zero reuse and x (260MB) > L2 (192MB): non-temporal both directions.
// grid = (124, 512): 124*256*4 = 126976 = MSPAT exactly.
// ---------------------------------------------------------------------------
__global__ __launch_bounds__(256) void k_scale(const float* __restrict__ x,
                                               const float* __restrict__ g,
                                               float* __restrict__ out) {
    const int bc = blockIdx.y;                       // b*64 + c
    const size_t base = (size_t)bc * MSPAT;
    const float gv = g[bc];
    const int i = (blockIdx.x * 256 + threadIdx.x) * 4;
    const v4f v = __builtin_nontemporal_load((const v4f*)(x + base + i));
    const v4f o = v * gv;
    __builtin_nontemporal_store(o, (v4f*)(out + base + i));
}

extern "C" void kernel_launch(void* const* d_in, const int* in_sizes, int n_in,
                              void* d_out, int out_size, void* d_ws, size_t ws_size,
                              hipStream_t stream) {
    const float* x   = (const float*)d_in[0];
    const float* w1  = (const float*)d_in[1];
    const float* b1  = (const float*)d_in[2];
    const float* w12 = (const float*)d_in[3];
    const float* b12 = (const float*)d_in[4];
    const float* w2  = (const float*)d_in[5];
    const float* b2  = (const float*)d_in[6];
    const float* w22 = (const float*)d_in[7];
    const float* b22 = (const float*)d_in[8];
    const float* w3  = (const float*)d_in[9];
    const float* b3  = (const float*)d_in[10];
    float* out = (float*)d_out;

    float* ws    = (float*)d_ws;
    float* pgram = ws;                                  // 8*62*4096
    float* psum  = ws + (size_t)NB * NKB * 4096;        // 8*62*64
    float* gbuf  = psum + (size_t)NB * NKB * 64;        // 8*64

    k_gram<<<dim3(NKB, NB), 256, 0, stream>>>(x, pgram, psum);
    k_small<<<NB, 256, 0, stream>>>(pgram, psum, w1, b1, w12, b12,
                                    w2, b2, w22, b22, w3, b3, gbuf);
    k_scale<<<dim3(124, NB * CC), 256, 0, stream>>>(x, gbuf, out);
}